// Router_18657337934009
// MI455X (gfx1250) — compile-verified
//
#include <hip/hip_runtime.h>
#include <hip/hip_bf16.h>

// ---------------------------------------------------------------------------
// MoE Router for MI455X (gfx1250, wave32).
//   T = 8192 tokens, D = 4096, E = 64 experts, top-k = 2, capacity = 320.
//   d_out layout (f32): [routing_weights T*E][dispatch T*E][router_loss 1]
// GEMM uses V_WMMA_F32_16X16X4_F32 (full f32 fidelity; the GEMM is HBM-bound
// streaming 134MB of x, so f32 matrix pipes are the right precision choice).
// ---------------------------------------------------------------------------

#define MOE_T 8192
#define MOE_D 4096
#define MOE_E 64
#define MOE_CAP 320u

typedef float v2f __attribute__((ext_vector_type(2)));
typedef float v8f __attribute__((ext_vector_type(8)));

// ---------------------------------------------------------------------------
// Kernel 0: zero the pass-0 expert histogram (ws is poisoned; re-zero per call)
// ---------------------------------------------------------------------------
__global__ void moe_zero_counts(unsigned int* __restrict__ counts) {
    counts[threadIdx.x] = 0u;
}

// ---------------------------------------------------------------------------
// Kernel 1: logits = x @ W^T via WMMA f32 16x16x4.
// Block = 256 threads = 8 waves; each wave computes a 16(token) x 64(expert)
// tile with 4 f32 accumulators, K = 4096 in steps of 4.
// A layout (16x4 f32):  lane = M + 16*(K>>1), vgpr = K&1  -> float2 per lane.
// B layout (4x16 f32):  lane = N + 16*(K>>1), vgpr = K&1  -> float2 per lane.
// C/D layout: vgpr j -> M = j + 8*(lane>>4), N = lane&15.
// ---------------------------------------------------------------------------
__global__ __launch_bounds__(256) void moe_gemm_logits(
    const float* __restrict__ x,   // [T, D]
    const float* __restrict__ W,   // [E, D]
    float* __restrict__ logits)    // [T, E]
{
    const int lane = threadIdx.x & 31;
    const int wave = threadIdx.x >> 5;
    const int half = lane >> 4;      // selects K pair (0 -> K 0,1 ; 1 -> K 2,3)
    const int l    = lane & 15;      // M index for A, N index for B

    const int tokBase = (blockIdx.x * 8 + wave) * 16;

    const float* aPtr = x + (size_t)(tokBase + l) * MOE_D + 2 * half;
    const float* bPtr = W + (size_t)l * MOE_D + 2 * half;   // expert row l

    v8f acc0 = {}; v8f acc1 = {}; v8f acc2 = {}; v8f acc3 = {};

#pragma unroll 4
    for (int k = 0; k < MOE_D; k += 4) {
        v2f a  = *(const v2f*)(aPtr + k);
        v2f b0 = *(const v2f*)(bPtr + k);
        v2f b1 = *(const v2f*)(bPtr + 16 * MOE_D + k);
        v2f b2 = *(const v2f*)(bPtr + 32 * MOE_D + k);
        v2f b3 = *(const v2f*)(bPtr + 48 * MOE_D + k);
        acc0 = __builtin_amdgcn_wmma_f32_16x16x4_f32(false, a, false, b0,
                                                     (short)0, acc0, false, false);
        acc1 = __builtin_amdgcn_wmma_f32_16x16x4_f32(false, a, false, b1,
                                                     (short)0, acc1, false, false);
        acc2 = __builtin_amdgcn_wmma_f32_16x16x4_f32(false, a, false, b2,
                                                     (short)0, acc2, false, false);
        acc3 = __builtin_amdgcn_wmma_f32_16x16x4_f32(false, a, false, b3,
                                                     (short)0, acc3, false, false);
    }

#pragma unroll
    for (int j = 0; j < 8; ++j) {
        const int row = tokBase + j + 8 * half;
        float* o = logits + (size_t)row * MOE_E + l;
        o[0]  = acc0[j];
        o[16] = acc1[j];
        o[32] = acc2[j];
        o[48] = acc3[j];
    }
}

// ---------------------------------------------------------------------------
// Kernel 2: per-token softmax over 64 experts (in place: logits -> weights),
// top-2 with renormalized weights, pass-0 histogram (mask always true since
// counts start at 0 < 320), and block-level partial sums of logits^2.
// ---------------------------------------------------------------------------
__global__ __launch_bounds__(256) void moe_softmax_topk(
    float* __restrict__ rw,              // in: logits, out: routing_weights
    float4* __restrict__ records,        // per token: {e0, e1, w0n, w1n}
    unsigned int* __restrict__ counts0,  // pass-0 histogram
    float* __restrict__ zpart)           // per-block sum(logits^2)
{
    const int t = blockIdx.x * 256 + threadIdx.x;

    float v[MOE_E];
    const float4* p = (const float4*)(rw + (size_t)t * MOE_E);
#pragma unroll
    for (int i = 0; i < 16; ++i) {
        float4 q = p[i];
        v[4 * i + 0] = q.x; v[4 * i + 1] = q.y;
        v[4 * i + 2] = q.z; v[4 * i + 3] = q.w;
    }

    float zsum = 0.f;
#pragma unroll
    for (int i = 0; i < MOE_E; ++i) zsum += v[i] * v[i];

    float m = v[0];
#pragma unroll
    for (int i = 1; i < MOE_E; ++i) m = fmaxf(m, v[i]);

    float s = 0.f;
#pragma unroll
    for (int i = 0; i < MOE_E; ++i) { float e = __expf(v[i] - m); v[i] = e; s += e; }
    const float inv = 1.f / s;

    // top-2 over exp values (monotonic in logits; ties keep lower index)
    float m1 = v[0], m2 = -1.f;
    int   i1 = 0,    i2 = 0;
#pragma unroll
    for (int i = 1; i < MOE_E; ++i) {
        float vi = v[i];
        if (vi > m1)      { m2 = m1; i2 = i1; m1 = vi; i1 = i; }
        else if (vi > m2) { m2 = vi; i2 = i; }
    }

    // write routing_weights (softmax) back in place
    float4* o = (float4*)(rw + (size_t)t * MOE_E);
#pragma unroll
    for (int i = 0; i < 16; ++i) {
        float4 q;
        q.x = v[4 * i + 0] * inv; q.y = v[4 * i + 1] * inv;
        q.z = v[4 * i + 2] * inv; q.w = v[4 * i + 3] * inv;
        o[i] = q;
    }

    const float w0 = m1 * inv, w1 = m2 * inv;
    const float sn = 1.f / (w0 + w1 + 1e-8f);
    float4 rec;
    rec.x = __int_as_float(i1);
    rec.y = __int_as_float(i2);
    rec.z = w0 * sn;
    rec.w = w1 * sn;
    records[t] = rec;

    atomicAdd(&counts0[i1], 1u);   // integer atomic: deterministic

    // deterministic tree reduction of zsum within the block
    __shared__ float red[256];
    red[threadIdx.x] = zsum;
    __syncthreads();
#pragma unroll
    for (int off = 128; off > 0; off >>= 1) {
        if (threadIdx.x < off) red[threadIdx.x] += red[threadIdx.x + off];
        __syncthreads();
    }
    if (threadIdx.x == 0) zpart[blockIdx.x] = red[0];
}

// ---------------------------------------------------------------------------
// Kernel 3: emit dispatch rows. Pass-1 mask = counts0[e1] < capacity, then
// row-normalize. d_out is poisoned, so the full 64-float row is written.
// ---------------------------------------------------------------------------
__global__ __launch_bounds__(256) void moe_dispatch_rows(
    const float4* __restrict__ records,
    const unsigned int* __restrict__ counts0,
    float* __restrict__ dispatch)        // [T, E]
{
    const int t = blockIdx.x * 256 + threadIdx.x;
    const float4 r = records[t];
    const int e0 = __float_as_int(r.x);
    const int e1 = __float_as_int(r.y);
    const bool mask = counts0[e1] < MOE_CAP;

    const float s   = r.z + (mask ? r.w : 0.f);
    const float inv = 1.f / (s + 1e-8f);

    float* row = dispatch + (size_t)t * MOE_E;
    float4* row4 = (float4*)row;
    const float4 z4 = make_float4(0.f, 0.f, 0.f, 0.f);
#pragma unroll
    for (int i = 0; i < 16; ++i) row4[i] = z4;   // stores from one lane stay ordered
    row[e0] = r.z * inv;
    if (mask) row[e1] = r.w * inv;
}

// ---------------------------------------------------------------------------
// Kernel 4: losses. 64 threads -> exact-order per-expert dispatch column sums
// (bit-deterministic, no float atomics), then thread 0 folds in z-loss.
// ---------------------------------------------------------------------------
__global__ __launch_bounds__(64) void moe_finalize(
    const float4* __restrict__ records,
    const unsigned int* __restrict__ counts0,
    const float* __restrict__ zpart,
    float* __restrict__ out_loss)
{
    const int e = threadIdx.x;   // 0..63
    float ec = 0.f;
    for (int t = 0; t < MOE_T; ++t) {
        const float4 r = records[t];
        const int e0 = __float_as_int(r.x);
        const int e1 = __float_as_int(r.y);
        const bool mask = counts0[e1] < MOE_CAP;
        const float s   = r.z + (mask ? r.w : 0.f);
        const float inv = 1.f / (s + 1e-8f);
        if (e0 == e)          ec += r.z * inv;
        if (mask && e1 == e)  ec += r.w * inv;
    }
    __shared__ float secs[MOE_E];
    secs[e] = ec;
    __syncthreads();
    if (e == 0) {
        const float invT   = 1.f / (float)MOE_T;
        const float target = (float)(MOE_T * 2 / MOE_E) * invT;   // 256/8192
        float lb = 0.f;
        for (int i = 0; i < MOE_E; ++i) {
            const float d = secs[i] * invT - target;
            lb += d * d;
        }
        lb *= (1.f / (float)MOE_E);
        float z = 0.f;
        for (int i = 0; i < 32; ++i) z += zpart[i];
        z *= (1.f / (float)(MOE_T * MOE_E));
        out_loss[0] = 0.001f * z + 0.001f * lb;
    }
}

// ---------------------------------------------------------------------------
// Launcher
// ---------------------------------------------------------------------------
extern "C" void kernel_launch(void* const* d_in, const int* in_sizes, int n_in,
                              void* d_out, int out_size, void* d_ws, size_t ws_size,
                              hipStream_t stream) {
    (void)in_sizes; (void)n_in; (void)out_size; (void)ws_size;

    const float* x = (const float*)d_in[0];   // [32*256, 4096]
    const float* W = (const float*)d_in[1];   // [64, 4096]

    float* out      = (float*)d_out;
    float* rw       = out;                       // routing_weights [T*E]
    float* dispatch = out + (size_t)MOE_T * MOE_E;
    float* loss     = out + 2 * (size_t)MOE_T * MOE_E;

    // workspace layout
    float4*       records = (float4*)d_ws;                                    // T*16B
    unsigned int* counts0 = (unsigned int*)((char*)d_ws + (size_t)MOE_T * 16);// 64*4B
    float*        zpart   = (float*)((char*)d_ws + (size_t)MOE_T * 16 + 256); // 32*4B

    moe_zero_counts<<<1, 64, 0, stream>>>(counts0);

    // 64 blocks x 8 waves x 16 tokens = 8192 tokens
    moe_gemm_logits<<<64, 256, 0, stream>>>(x, W, rw);

    moe_softmax_topk<<<32, 256, 0, stream>>>(rw, records, counts0, zpart);

    moe_dispatch_rows<<<32, 256, 0, stream>>>(records, counts0, dispatch);

    moe_finalize<<<1, 64, 0, stream>>>(records, counts0, zpart, loss);
}